// PhotonicNeuralNetwork_85933705658690
// MI455X (gfx1250) — compile-verified
//
#include <hip/hip_runtime.h>

// ---------------------------------------------------------------------------
// 6-qubit real state-vector simulator, restructured as:
//   out[e] = f( E_rows , ROT3(x_e) * Msub * v(x_e) )
// Msub (64x8, x-independent middle block incl. real-ified CCC-Y) and rows 0/7
// of E (8x8 final entangler) are built per block; the (16x8)x(8x64) per-wave
// GEMM runs on V_WMMA_F32_16X16X4_F32.
//
// Qubit q <-> bit (5-q) of the 64-state linear index (q0 is the MSB, matching
// the reference's (B,2,2,2,2,2,2) layout). i = idx>>3 = q0q1q2, j = idx&7.
// All trig uses the hardware fast path (__sinf/__cosf): half-angles are
// bounded to ~|12| rad, well inside v_sin_f32/v_cos_f32 accuracy range.
// ---------------------------------------------------------------------------

typedef __attribute__((ext_vector_type(2))) float v2f;
typedef __attribute__((ext_vector_type(8))) float v8f;

#define NSTATE 64

// (controlled) RY on the 64-dim real state; 'half' is theta/2.
__device__ inline void cry64(float* st, int tmask, int cmask, float half) {
    float c = __cosf(half), s = __sinf(half);
    #pragma unroll 1
    for (int i0 = 0; i0 < NSTATE; ++i0) {
        if ((i0 & tmask) == 0 && (i0 & cmask) == cmask) {
            int i1 = i0 | tmask;
            float a0 = st[i0], a1 = st[i1];
            st[i0] = c * a0 - s * a1;
            st[i1] = s * a0 + c * a1;
        }
    }
}

// Pauli-Y with the global i factor dropped (valid: uniform phase inside the
// q3q4q5=111 subspace, never mixed with other subspaces afterwards).
__device__ inline void y64_real(float* st, int tmask, int cmask) {
    #pragma unroll 1
    for (int i0 = 0; i0 < NSTATE; ++i0) {
        if ((i0 & tmask) == 0 && (i0 & cmask) == cmask) {
            int i1 = i0 | tmask;
            float a0 = st[i0], a1 = st[i1];
            st[i0] = -a1;
            st[i1] = a0;
        }
    }
}

__device__ inline void cry8(float* st, int tmask, int cmask, float half) {
    float c = __cosf(half), s = __sinf(half);
    #pragma unroll 1
    for (int i0 = 0; i0 < 8; ++i0) {
        if ((i0 & tmask) == 0 && (i0 & cmask) == cmask) {
            int i1 = i0 | tmask;
            float a0 = st[i0], a1 = st[i1];
            st[i0] = c * a0 - s * a1;
            st[i1] = s * a0 + c * a1;
        }
    }
}

__global__ __launch_bounds__(128) void qsim_kernel(const float* __restrict__ x,
                                                   const float* __restrict__ p,
                                                   float* __restrict__ out, int B) {
    __shared__ float sM[NSTATE * 8];   // Msub[n][k], n-major
    __shared__ float sR[16];           // E row0 (0..7), E row7 (8..15)
    __shared__ float sW[4][32][65];    // per-wave WMMA result staging, padded

    const int tid  = threadIdx.x;
    const int lane = tid & 31;
    const int wave = tid >> 5;

    // ---- per-block constant precompute (x-independent, tiny) --------------
    if (tid < 8) {
        // column k=tid of the middle-block operator, input basis |q0q1q2=k, 000>
        float st[NSTATE];
        #pragma unroll 1
        for (int i = 0; i < NSTATE; ++i) st[i] = 0.f;
        st[tid * 8] = 1.f;
        // masks: q0=32 q1=16 q2=8 q3=4 q4=2 q5=1
        cry64(st, 4, 8,  0.5f * p[3]);  cry64(st, 4, 16, 0.5f * p[4]);
        cry64(st, 4, 32, 0.5f * p[5]);  cry64(st, 4, 0,  0.5f * p[6]);
        cry64(st, 2, 8,  0.5f * p[7]);  cry64(st, 2, 16, 0.5f * p[8]);
        cry64(st, 2, 32, 0.5f * p[9]);  cry64(st, 2, 0,  0.5f * p[10]);
        cry64(st, 1, 8,  0.5f * p[11]); cry64(st, 1, 16, 0.5f * p[12]);
        cry64(st, 1, 32, 0.5f * p[13]); cry64(st, 1, 0,  0.5f * p[14]);
        cry64(st, 4, 0, 0.25f); cry64(st, 2, 0, 0.25f); cry64(st, 1, 0, 0.25f);
        cry64(st, 2, 4, 0.5f * p[15]);  cry64(st, 1, 4, 0.5f * p[16]);
        cry64(st, 4, 2, 0.5f * p[17]);  cry64(st, 1, 2, 0.5f * p[18]);
        cry64(st, 4, 1, 0.5f * p[19]);  cry64(st, 2, 1, 0.5f * p[20]);
        cry64(st, 4, 0, 0.5f * p[21]);  cry64(st, 2, 0, 0.5f * p[22]);
        cry64(st, 1, 0, 0.5f * p[23]);
        y64_real(st, 32, 7); y64_real(st, 16, 7); y64_real(st, 8, 7);
        #pragma unroll 1
        for (int n = 0; n < NSTATE; ++n) sM[n * 8 + tid] = st[n];
    } else if (tid < 16) {
        // rows 0 and 7 of the final 0-1-2 entangler E (8x8, q0=bit2)
        int col = tid - 8;
        float st[8];
        #pragma unroll
        for (int i = 0; i < 8; ++i) st[i] = (i == col) ? 1.f : 0.f;
        cry8(st, 2, 4, 0.5f * p[27]); cry8(st, 1, 4, 0.5f * p[28]);
        cry8(st, 4, 2, 0.5f * p[29]); cry8(st, 1, 2, 0.5f * p[30]);
        cry8(st, 4, 1, 0.5f * p[31]); cry8(st, 2, 1, 0.5f * p[32]);
        cry8(st, 4, 0, 0.5f * p[33]); cry8(st, 2, 0, 0.5f * p[34]);
        cry8(st, 1, 0, 0.5f * p[35]);
        sR[col]     = st[0];   // E[0][col]
        sR[8 + col] = st[7];   // E[7][col]
    }
    __syncthreads();

    // ---- WMMA GEMM: W(16x64) = V(16x8) * Msub^T(8x64), per wave, 2 M-tiles --
    const int row = lane & 15;   // = M for A, = N for B/C
    const int kh  = lane >> 4;   // lane-half selects K pair (ISA 16x4 f32 layout)
    const int waveBase = blockIdx.x * 128 + wave * 32;

    // B operand (Msub^T), layout mirrors A: vgpr0 <-> K=0/2, vgpr1 <-> K=1/3
    v2f Bop[2][4];
    #pragma unroll
    for (int kc = 0; kc < 2; ++kc) {
        #pragma unroll
        for (int nt = 0; nt < 4; ++nt) {
            int n  = row + 16 * nt;
            int kb = kc * 4 + 2 * kh;
            v2f b; b.x = sM[n * 8 + kb]; b.y = sM[n * 8 + kb + 1];
            Bop[kc][nt] = b;
        }
    }

    #pragma unroll
    for (int mt = 0; mt < 2; ++mt) {
        int e  = waveBase + mt * 16 + row;
        int ec = e < B ? e : (B - 1);
        float xv = x[ec];
        // product state v(x): composed angles (3x+p0)/2, (9x+p1)/2, (x+p2)/2
        float t0 = 0.5f * (3.f * xv + p[0]);
        float t1 = 0.5f * (9.f * xv + p[1]);
        float t2 = 0.5f * (xv + p[2]);
        float c0 = __cosf(t0), s0 = __sinf(t0);
        float c1 = __cosf(t1), s1 = __sinf(t1);
        float c2 = __cosf(t2), s2 = __sinf(t2);
        // A slot k-indices are {2kh, 2kh+1, 4+2kh, 5+2kh}: bit2 = chunk,
        // bit1 = kh (only lane-half-dependent factor), bit0 = slot parity.
        float m1 = kh ? s1 : c1;              // single select, no vk[] ladder
        float e0c = c0 * m1, e0s = s0 * m1;
        v2f A0; A0.x = e0c * c2; A0.y = e0c * s2;   // K chunk 0..3
        v2f A1; A1.x = e0s * c2; A1.y = e0s * s2;   // K chunk 4..7

        #pragma unroll
        for (int nt = 0; nt < 4; ++nt) {
            v8f acc = {0.f, 0.f, 0.f, 0.f, 0.f, 0.f, 0.f, 0.f};
            acc = __builtin_amdgcn_wmma_f32_16x16x4_f32(
                false, A0, false, Bop[0][nt], (short)0, acc, false, false);
            acc = __builtin_amdgcn_wmma_f32_16x16x4_f32(
                false, A1, false, Bop[1][nt], (short)0, acc, false, false);
            // C/D layout: vgpr r, lane L -> M = r + 8*(L>>4), N = (L&15)+16*nt
            #pragma unroll
            for (int r = 0; r < 8; ++r)
                sW[wave][mt * 16 + r + 8 * kh][row + 16 * nt] = acc[r];
        }
    }
    __syncthreads();

    // ---- per-element tail: one lane owns one element ----------------------
    int e  = waveBase + lane;
    int ec = e < B ? e : (B - 1);
    float w[NSTATE];
    #pragma unroll
    for (int n = 0; n < NSTATE; ++n) w[n] = sW[wave][lane][n];

    float xv = x[ec];
    // composed encoding-2 rotations: (7x+p24-0.5)/2 on q0, etc.
    const float hh[3] = {0.5f * (7.f * xv + p[24] - 0.5f),
                         0.5f * (17.f * xv + p[25] - 0.5f),
                         0.5f * (xv + p[26] - 0.5f)};
    const int tm[3] = {32, 16, 8};
    #pragma unroll
    for (int g = 0; g < 3; ++g) {
        float c = __cosf(hh[g]), s = __sinf(hh[g]);
        int m = tm[g];
        #pragma unroll
        for (int i = 0; i < NSTATE; ++i) {
            if (!(i & m)) {
                float a0 = w[i], a1 = w[i | m];
                w[i]     = c * a0 - s * a1;
                w[i | m] = s * a0 + c * a1;
            }
        }
    }

    // p000/p111 via rows 0,7 of E: d = E_row . w[:,j], sum d^2 over j
    float p000 = 0.f, p111 = 0.f;
    #pragma unroll
    for (int j = 0; j < 8; ++j) {
        float d0 = 0.f, d7 = 0.f;
        #pragma unroll
        for (int i = 0; i < 8; ++i) {
            float wv = w[i * 8 + j];
            d0 += sR[i] * wv;
            d7 += sR[8 + i] * wv;
        }
        p000 += d0 * d0;
        p111 += d7 * d7;
    }
    if (e < B) out[e] = 2.f * (p111 - p000);
}

extern "C" void kernel_launch(void* const* d_in, const int* in_sizes, int n_in,
                              void* d_out, int out_size, void* d_ws, size_t ws_size,
                              hipStream_t stream) {
    const float* x = (const float*)d_in[0];
    const float* p = (const float*)d_in[1];
    float* out = (float*)d_out;
    int B = in_sizes[0];
    int blocks = (B + 127) / 128;   // 128 elements per 128-thread block (4 waves x 32)
    qsim_kernel<<<blocks, 128, 0, stream>>>(x, p, out, B);
}